// SinkhornOTLoss_57019985821814
// MI455X (gfx1250) — compile-verified
//
#include <hip/hip_runtime.h>
#include <hip/hip_bf16.h>
#include <math.h>

#define EPS        0.1f
#define THRESH     0.001f
#define N_SRC      1024
#define M_TGT      16384
#define LOG2_M     14
#define DDIM       768
#define MAX_ITERS  100
#define NPARTIAL   4096

typedef float v2f __attribute__((ext_vector_type(2)));
typedef float v4f __attribute__((ext_vector_type(4)));
typedef float v8f __attribute__((ext_vector_type(8)));

// ---------------------------------------------------------------------------
// Row squared-norms: one block per row, 256 threads.
// ---------------------------------------------------------------------------
__global__ void sk_row_norms(const float* __restrict__ X, float* __restrict__ out, int d) {
    __shared__ float red[256];
    const int row = blockIdx.x;
    const float* xr = X + (size_t)row * d;
    float s = 0.f;
    for (int k = threadIdx.x; k < d; k += 256) {
        float x = xr[k];
        s += x * x;
    }
    red[threadIdx.x] = s;
    __syncthreads();
    for (int off = 128; off > 0; off >>= 1) {
        if (threadIdx.x < off) red[threadIdx.x] += red[threadIdx.x + off];
        __syncthreads();
    }
    if (threadIdx.x == 0) out[row] = red[0];
}

// ---------------------------------------------------------------------------
// GEMM + exp epilogue: K[i,j] = exp((2*src.tgt - |src|^2 - |tgt|^2)/EPS)
// Each wave computes a 16(M) x 64(N) strip with V_WMMA_F32_16X16X4_F32:
// 4 accumulators share one A-fragment per k-step -> 5 loads per 4 WMMA,
// and the 4 independent D->C chains give the WMMA pipe its ILP.
// Block = 128 threads (4 waves) covering 16 rows x 256 cols.
//
// A-frag (16x4, 2 VGPRs): lane L holds row M=L&15; K-offset=(L>=16)?2:0,
//   elements K=koff,koff+1 -> contiguous float2 from src (row-major).
// B-frag (4x16, 2 VGPRs): lane L holds col N=L&15; same K split
//   -> contiguous float2 from tgt (row-major) since B[k][j] = tgt[j][k].
// D/C (16x16, 8 VGPRs): elem g of lane L is M = g + ((L>=16)?8:0), N = L&15.
// ---------------------------------------------------------------------------
__global__ void sk_gemm_expK(const float* __restrict__ src, const float* __restrict__ tgt,
                             const float* __restrict__ sn, const float* __restrict__ tn,
                             float* __restrict__ K) {
    const int lane = threadIdx.x & 31;
    const int wave = threadIdx.x >> 5;
    const int i0 = blockIdx.y << 4;
    const int j0 = (blockIdx.x << 8) + (wave << 6);   // 64 columns per wave
    const int r    = lane & 15;
    const int koff = (lane >> 4) << 1;

    const float* aptr = src + (size_t)(i0 + r) * DDIM + koff;
    const float* bptr = tgt + (size_t)(j0 + r) * DDIM + koff;

    v8f c0 = {0.f,0.f,0.f,0.f,0.f,0.f,0.f,0.f};
    v8f c1 = c0, c2 = c0, c3 = c0;

    #pragma unroll 4
    for (int k = 0; k < DDIM; k += 4) {
        v2f a  = *(const v2f*)(aptr + k);
        v2f b0 = *(const v2f*)(bptr + k);
        v2f b1 = *(const v2f*)(bptr + k + 16 * DDIM);
        v2f b2 = *(const v2f*)(bptr + k + 32 * DDIM);
        v2f b3 = *(const v2f*)(bptr + k + 48 * DDIM);
        c0 = __builtin_amdgcn_wmma_f32_16x16x4_f32(false, a, false, b0, (short)0, c0, false, false);
        c1 = __builtin_amdgcn_wmma_f32_16x16x4_f32(false, a, false, b1, (short)0, c1, false, false);
        c2 = __builtin_amdgcn_wmma_f32_16x16x4_f32(false, a, false, b2, (short)0, c2, false, false);
        c3 = __builtin_amdgcn_wmma_f32_16x16x4_f32(false, a, false, b3, (short)0, c3, false, false);
    }

    const int ibase = i0 + ((lane >> 4) << 3);
    float snv[8];
    #pragma unroll
    for (int g = 0; g < 8; ++g) snv[g] = sn[ibase + g];

#define SK_EPILOG(CC, T)                                                            \
    {                                                                               \
        const int j = j0 + ((T) << 4) + r;                                          \
        const float tnj = tn[j];                                                    \
        _Pragma("unroll")                                                           \
        for (int g = 0; g < 8; ++g) {                                               \
            const int i = ibase + g;                                                \
            const float e = (2.0f * (CC)[g] - snv[g] - tnj) * (1.0f / EPS);         \
            K[((size_t)i << LOG2_M) + j] = __expf(e);                               \
        }                                                                           \
    }
    SK_EPILOG(c0, 0)
    SK_EPILOG(c1, 1)
    SK_EPILOG(c2, 2)
    SK_EPILOG(c3, 3)
#undef SK_EPILOG
}

// ---------------------------------------------------------------------------
// Init u = 1, v = 1, done = 0, err = 0
// ---------------------------------------------------------------------------
__global__ void sk_init(float* __restrict__ u, float* __restrict__ v,
                        unsigned* __restrict__ err_bits, int* __restrict__ done) {
    const int t = blockIdx.x * blockDim.x + threadIdx.x;
    if (t < N_SRC) u[t] = 1.f;
    if (t < M_TGT) v[t] = 1.f;
    if (t == 0) { *err_bits = 0u; *done = 0; }
}

// ---------------------------------------------------------------------------
// u_new[i] = a / (K[i,:] . v + 1e-10); float4 row dot, one block per row.
// Track max|u_new - u| via atomicMax on float bits (diffs are >= 0).
// ---------------------------------------------------------------------------
__global__ void sk_u_update(const float* __restrict__ K, const float* __restrict__ v,
                            float* __restrict__ u, float a_val,
                            unsigned* __restrict__ err_bits, const int* __restrict__ done) {
    if (*done) return;
    __shared__ float red[256];
    const int i = blockIdx.x;
    const v4f* Kr = (const v4f*)(K + ((size_t)i << LOG2_M));
    const v4f* vv = (const v4f*)v;
    float s = 0.f;
    for (int q = threadIdx.x; q < (M_TGT >> 2); q += 256) {
        const v4f kk = Kr[q];
        const v4f vq = vv[q];
        s += kk[0] * vq[0] + kk[1] * vq[1] + kk[2] * vq[2] + kk[3] * vq[3];
    }
    red[threadIdx.x] = s;
    __syncthreads();
    for (int off = 128; off > 0; off >>= 1) {
        if (threadIdx.x < off) red[threadIdx.x] += red[threadIdx.x + off];
        __syncthreads();
    }
    if (threadIdx.x == 0) {
        const float un   = a_val / (red[0] + 1e-10f);
        const float diff = fabsf(un - u[i]);
        u[i] = un;
        atomicMax(err_bits, __float_as_uint(diff));
    }
}

// ---------------------------------------------------------------------------
// v[j] = b / (K[:,j] . u + 1e-10); each thread owns 4 consecutive columns
// (b128 row loads, coalesced), u[i] is a uniform scalar broadcast.
// ---------------------------------------------------------------------------
__global__ void sk_v_update(const float* __restrict__ K, const float* __restrict__ u,
                            float* __restrict__ v, float b_val,
                            const int* __restrict__ done) {
    if (*done) return;
    const int j4 = (blockIdx.x * blockDim.x + threadIdx.x) << 2;
    float s0 = 0.f, s1 = 0.f, s2 = 0.f, s3 = 0.f;
    #pragma unroll 4
    for (int i = 0; i < N_SRC; ++i) {
        const v4f kk = *(const v4f*)(K + ((size_t)i << LOG2_M) + j4);
        const float ui = u[i];
        s0 += kk[0] * ui;
        s1 += kk[1] * ui;
        s2 += kk[2] * ui;
        s3 += kk[3] * ui;
    }
    v4f r;
    r[0] = b_val / (s0 + 1e-10f);
    r[1] = b_val / (s1 + 1e-10f);
    r[2] = b_val / (s2 + 1e-10f);
    r[3] = b_val / (s3 + 1e-10f);
    *(v4f*)(v + j4) = r;
}

// ---------------------------------------------------------------------------
// After both updates: done |= (err < THRESH); reset err for next iteration.
// Matches the reference's "freeze after the converging update" semantics.
// ---------------------------------------------------------------------------
__global__ void sk_done_update(unsigned* __restrict__ err_bits, int* __restrict__ done) {
    if (!*done && __uint_as_float(*err_bits) < THRESH) *done = 1;
    *err_bits = 0u;
}

// ---------------------------------------------------------------------------
// Partial sums of u_i * K_ij * v_j * cost_ij with cost = -EPS*ln(K) (exact
// inverse of the stored transform; guarded against K underflow to 0).
// float4 loads; deterministic fixed-stride mapping + tree reduction.
// ---------------------------------------------------------------------------
__global__ void sk_pkc_partial(const float* __restrict__ K, const float* __restrict__ u,
                               const float* __restrict__ v, float* __restrict__ partial) {
    __shared__ float red[256];
    const size_t total4 = ((size_t)N_SRC << LOG2_M) >> 2;
    const size_t stride = (size_t)gridDim.x * blockDim.x;
    float s = 0.f;
    for (size_t q = (size_t)blockIdx.x * blockDim.x + threadIdx.x; q < total4; q += stride) {
        const size_t idx = q << 2;
        const int i = (int)(idx >> LOG2_M);
        const int j = (int)(idx & (M_TGT - 1));
        const v4f kk = *(const v4f*)(K + idx);
        const v4f vv = *(const v4f*)(v + j);
        const float ui = u[i];
        #pragma unroll
        for (int e = 0; e < 4; ++e) {
            const float kv   = kk[e];
            const float cost = (kv > 0.f) ? (-EPS * __logf(kv)) : 0.f;
            s += ui * kv * vv[e] * cost;
        }
    }
    red[threadIdx.x] = s;
    __syncthreads();
    for (int off = 128; off > 0; off >>= 1) {
        if (threadIdx.x < off) red[threadIdx.x] += red[threadIdx.x + off];
        __syncthreads();
    }
    if (threadIdx.x == 0) partial[blockIdx.x] = red[0];
}

__global__ void sk_finalize(const float* __restrict__ partial, float* __restrict__ out) {
    __shared__ float red[256];
    float s = 0.f;
    for (int t = threadIdx.x; t < NPARTIAL; t += 256) s += partial[t];
    red[threadIdx.x] = s;
    __syncthreads();
    for (int off = 128; off > 0; off >>= 1) {
        if (threadIdx.x < off) red[threadIdx.x] += red[threadIdx.x + off];
        __syncthreads();
    }
    if (threadIdx.x == 0) {
        const float d = red[0] / (float)(N_SRC + M_TGT);
        out[0] = sqrtf(d + 1e-8f);
    }
}

// ---------------------------------------------------------------------------
extern "C" void kernel_launch(void* const* d_in, const int* in_sizes, int n_in,
                              void* d_out, int out_size, void* d_ws, size_t ws_size,
                              hipStream_t stream) {
    const float* src = (const float*)d_in[0];  // (1024, 768)
    const float* tgt = (const float*)d_in[1];  // (16384, 768)

    // workspace layout (256-byte aligned slices)
    char* ws = (char*)d_ws;
    size_t off = 0;
    auto alloc = [&](size_t bytes) -> char* {
        char* p = ws + off;
        off = (off + bytes + 255) & ~(size_t)255;
        return p;
    };
    float*    K        = (float*)alloc((size_t)N_SRC * M_TGT * sizeof(float)); // 64 MB
    float*    sn       = (float*)alloc(N_SRC * sizeof(float));
    float*    tn       = (float*)alloc(M_TGT * sizeof(float));
    float*    u        = (float*)alloc(N_SRC * sizeof(float));
    float*    v        = (float*)alloc(M_TGT * sizeof(float));
    float*    partial  = (float*)alloc(NPARTIAL * sizeof(float));
    unsigned* err_bits = (unsigned*)alloc(sizeof(unsigned));
    int*      done     = (int*)alloc(sizeof(int));
    (void)ws_size; (void)in_sizes; (void)n_in; (void)out_size;

    // 1) row norms
    sk_row_norms<<<N_SRC, 256, 0, stream>>>(src, sn, DDIM);
    sk_row_norms<<<M_TGT, 256, 0, stream>>>(tgt, tn, DDIM);

    // 2) K = exp(-cost/eps) via f32 WMMA GEMM (16x64 strip per wave)
    sk_gemm_expK<<<dim3(M_TGT / 256, N_SRC / 16), 128, 0, stream>>>(src, tgt, sn, tn, K);

    // 3) Sinkhorn iterations (K is L2-resident: 64 MB < 192 MB L2)
    sk_init<<<M_TGT / 256, 256, 0, stream>>>(u, v, err_bits, done);
    const float a_val = 1.0f / N_SRC;
    const float b_val = 1.0f / M_TGT;
    for (int it = 0; it < MAX_ITERS; ++it) {
        sk_u_update<<<N_SRC, 256, 0, stream>>>(K, v, u, a_val, err_bits, done);
        sk_v_update<<<M_TGT / 1024, 256, 0, stream>>>(K, u, v, b_val, done);
        sk_done_update<<<1, 1, 0, stream>>>(err_bits, done);
    }

    // 4) <P, C> reduction + sqrt
    sk_pkc_partial<<<NPARTIAL, 256, 0, stream>>>(K, u, v, partial);
    sk_finalize<<<1, 256, 0, stream>>>(partial, (float*)d_out);
}